// BCE_for_non_zero_37220186587139
// MI455X (gfx1250) — compile-verified
//
#include <hip/hip_runtime.h>

#define NCH 14
#define NB  1024
#define TPB 256

typedef float v2f __attribute__((ext_vector_type(2)));
typedef float v8f __attribute__((ext_vector_type(8)));

__device__ __forceinline__ float row_loss(const float* xv, const float* tv, const int* g) {
    // per-row group sums of targets (group 0 never needs its sum)
    float s1 = 0.f, s2 = 0.f, s3 = 0.f;
#pragma unroll
    for (int c = 0; c < NCH; ++c) {
        float t = tv[c];
        s1 += (g[c] == 1) ? t : 0.f;
        s2 += (g[c] == 2) ? t : 0.f;
        s3 += (g[c] == 3) ? t : 0.f;
    }
    float acc = 0.f;
#pragma unroll
    for (int c = 0; c < NCH; ++c) {
        float x = xv[c], t = tv[c];
        // BCE with logits: max(x,0) - x*t + log1p(exp(-|x|))
        float e   = __expf(-fabsf(x));
        float bce = fmaxf(x, 0.f) - x * t + __logf(1.f + e);
        float sg  = (g[c] == 1) ? s1 : ((g[c] == 2) ? s2 : s3);
        float m   = (g[c] == 0 || sg > 0.f) ? 1.f : 0.f;
        acc += bce * m;
    }
    return acc;
}

// Stage A: stream 224MB, one partial per block (no atomics -> deterministic).
__global__ void bce_partials(const float* __restrict__ x, const float* __restrict__ t,
                             const int* __restrict__ grp, float* __restrict__ partials,
                             long long nsuper) {
    int g[NCH];
#pragma unroll
    for (int c = 0; c < NCH; ++c) g[c] = grp[c];  // uniform -> s_load

    long long tid    = (long long)blockIdx.x * blockDim.x + threadIdx.x;
    long long stride = (long long)gridDim.x * blockDim.x;
    float acc = 0.f;
    for (long long i = tid; i < nsuper; i += stride) {
        // 2 rows = 28 floats = 7 x float4 per tensor, 16B aligned (i*112)
        const float4* xp = reinterpret_cast<const float4*>(x) + i * 7;
        const float4* tp = reinterpret_cast<const float4*>(t) + i * 7;
        float xv[28], tv[28];
#pragma unroll
        for (int k = 0; k < 7; ++k) {
            float4 a = xp[k];
            float4 b = tp[k];
            xv[4 * k] = a.x; xv[4 * k + 1] = a.y; xv[4 * k + 2] = a.z; xv[4 * k + 3] = a.w;
            tv[4 * k] = b.x; tv[4 * k + 1] = b.y; tv[4 * k + 2] = b.z; tv[4 * k + 3] = b.w;
        }
        acc += row_loss(xv, tv, g);
        acc += row_loss(xv + NCH, tv + NCH, g);
    }

    // wave32 reduction
#pragma unroll
    for (int m = 16; m >= 1; m >>= 1) acc += __shfl_xor(acc, m, 32);

    __shared__ float ls[TPB / 32];
    int lane = threadIdx.x & 31, wid = threadIdx.x >> 5;
    if (lane == 0) ls[wid] = acc;
    __syncthreads();
    if (threadIdx.x == 0) {
        float s = 0.f;
#pragma unroll
        for (int w = 0; w < TPB / 32; ++w) s += ls[w];
        partials[blockIdx.x] = s;
    }
}

// Stage B: one wave. Lane-strided sum of block partials, then a 3-round
// v_wmma_f32_16x16x4_f32 reduction tree (A = ones => D[m,n] = sum_k B[k,n]).
__global__ void bce_finalize(const float* __restrict__ partials, int npart,
                             float* __restrict__ out, float inv_count) {
    int l = threadIdx.x;  // 0..31, EXEC all ones (single full wave)
    float p = 0.f;
    for (int i = l; i < npart; i += 32) p += partials[i];

    v2f ones; ones.x = 1.f; ones.y = 1.f;
    v8f zero = {};

    // Round 1: 32 -> 16. B row K0 = lanes 0..15 of b.x, row K2 = lanes 16..31.
    // D[m,n] = p[n] + p[n+16] = q[n] for all m; d1[0] holds q[l%16] on every lane.
    v2f b1; b1.x = p; b1.y = 0.f;
    v8f d1 = __builtin_amdgcn_wmma_f32_16x16x4_f32(false, ones, false, b1,
                                                   (short)0, zero, false, false);
    float v1 = d1[0];

    int n   = l & 15;
    int off = (l < 16) ? 0 : 2;

    // Round 2: 16 -> 4. Column n's four K slots live at (b.x,b.y) of lanes n and n+16.
    // Stage q[4n..4n+3] there; only columns n<4 are meaningful.
    v2f b2;
    b2.x = __shfl(v1, (4 * n + off) & 15, 32);
    b2.y = __shfl(v1, (4 * n + off + 1) & 15, 32);
    v8f d2 = __builtin_amdgcn_wmma_f32_16x16x4_f32(false, ones, false, b2,
                                                   (short)0, zero, false, false);
    float v2 = d2[0];  // Q[l%16], valid for l%16 < 4

    // Round 3: 4 -> 1. Column 0 gets Q[0..3]: lane0 {Q0,Q1}, lane16 {Q2,Q3}.
    v2f b3;
    b3.x = __shfl(v2, off, 32);
    b3.y = __shfl(v2, off + 1, 32);
    v8f d3 = __builtin_amdgcn_wmma_f32_16x16x4_f32(false, ones, false, b3,
                                                   (short)0, zero, false, false);

    if (l == 0) out[0] = d3[0] * inv_count;  // D[0,0] = total
}

extern "C" void kernel_launch(void* const* d_in, const int* in_sizes, int n_in,
                              void* d_out, int out_size, void* d_ws, size_t ws_size,
                              hipStream_t stream) {
    const float* x   = (const float*)d_in[0];
    const float* t   = (const float*)d_in[1];
    const int*   grp = (const int*)d_in[2];
    float*       out = (float*)d_out;
    float*       ws  = (float*)d_ws;  // NB block partials (4 KB)

    long long total  = (long long)in_sizes[0];   // B*C = 28,000,000
    long long nrows  = total / NCH;              // B = 2,000,000 (even)
    long long nsuper = nrows / 2;                // 2 rows per thread-iteration
    float inv_count  = 1.0f / (float)total;

    bce_partials<<<NB, TPB, 0, stream>>>(x, t, grp, ws, nsuper);
    bce_finalize<<<1, 32, 0, stream>>>(ws, NB, out, inv_count);
}